// SpLinear_13675175870903
// MI455X (gfx1250) — compile-verified
//
#include <hip/hip_runtime.h>
#include <hip/hip_bf16.h>
#include <stdint.h>

// ---------------------------------------------------------------------------
// 2:4 structured-sparse linear  y = x @ (mask*W)^T + b  on gfx1250 (MI455X)
//   x:    [4096, 4096]  fp32  (B*S flattened)
//   W:    [16384, 4096] fp32, 2:4 sparse along K (via mask)
//   out:  [4096, 16384] fp32
// Strategy: compress W -> bf16 2:4 packed values + 2-bit indices, convert x to
// bf16, then V_SWMMAC_F32_16X16X64_BF16 main loop (wave32).
// Wave tile 64(o) x 32(m): 4 sparse A frags x 2 dense B frags -> 8 swmmac/iter.
// Block = 8 waves (2o x 4m) -> 128 x 128 block tile. Output stores are
// non-temporal so the L2 (192 MB) stays dedicated to the hot W/x streams.
// ---------------------------------------------------------------------------

typedef __attribute__((ext_vector_type(16))) __bf16 v16bf;   // sparse A frag (16x32 stored)
typedef __attribute__((ext_vector_type(32))) __bf16 v32bf;   // dense  B frag (64x16)
typedef __attribute__((ext_vector_type(8)))  float  v8f;     // C/D frag (16x16 f32)
typedef __attribute__((ext_vector_type(4)))  float  v4f;     // native float4 for NT stores

#define IN_F   4096
#define OUT_F  16384
#define MROWS  4096            // BATCH * SEQ
#define KSTOR  (IN_F / 2)      // stored bf16 per weight row after 2:4 compression
#define KIDXB  (IN_F / 8)      // index bytes per row (4 bits per group-of-4)

// ---------------------------------------------------------------------------
// Kernel 1: fold mask into W; emit packed bf16 non-zeros + 2-bit index pairs.
// One thread handles 2 groups of 4 (8 original K) -> 4 bf16 values + 1 byte.
// Index nibble for group i sits at bits [4i+3:4i] of the lane's index dword,
// encoded {idx1[1:0], idx0[1:0]} with idx0 < idx1 (ISA 7.12.4 requirement).
// ---------------------------------------------------------------------------
__global__ __launch_bounds__(256) void k_compress_w(
    const float* __restrict__ w, const uint8_t* __restrict__ mask,
    __bf16* __restrict__ wv, uint8_t* __restrict__ widx)
{
    const long t  = (long)blockIdx.x * blockDim.x + threadIdx.x;   // OUT_F*KIDXB
    const int  o  = (int)(t >> 9);
    const int  bg = (int)(t & 511);
    const float*   wr = w    + (long)o * IN_F + bg * 8;
    const uint8_t* mr = mask + (long)o * IN_F + bg * 8;

    union { __bf16 h[4]; uint2 u; } vals;
    unsigned code = 0;
#pragma unroll
    for (int g = 0; g < 2; ++g) {
        int i0 = -1, i1 = -1;
#pragma unroll
        for (int j = 0; j < 4; ++j) {
            if (mr[g * 4 + j]) { if (i0 < 0) i0 = j; else i1 = j; }
        }
        if (i0 < 0) i0 = 0;                    // defensive (mask keeps exactly 2)
        if (i1 < 0) i1 = (i0 < 3) ? 3 : 2;
        vals.h[g * 2 + 0] = (__bf16)wr[g * 4 + i0];
        vals.h[g * 2 + 1] = (__bf16)wr[g * 4 + i1];
        code |= (unsigned)(i0 | (i1 << 2)) << (g * 4);
    }
    *reinterpret_cast<uint2*>(wv + (long)o * KSTOR + bg * 4) = vals.u;
    widx[(long)o * KIDXB + bg] = (uint8_t)code;
}

// ---------------------------------------------------------------------------
// Kernel 2: x fp32 -> bf16, 8 elements / thread, 16B packed stores.
// ---------------------------------------------------------------------------
__global__ __launch_bounds__(256) void k_cvt_x(
    const float* __restrict__ x, __bf16* __restrict__ xb, long n)
{
    const long i = ((long)blockIdx.x * blockDim.x + threadIdx.x) * 8;
    if (i >= n) return;
    v4f a = *reinterpret_cast<const v4f*>(x + i);
    v4f b = *reinterpret_cast<const v4f*>(x + i + 4);
    union { __bf16 h[8]; uint4 u; } p;
    p.h[0] = (__bf16)a.x; p.h[1] = (__bf16)a.y; p.h[2] = (__bf16)a.z; p.h[3] = (__bf16)a.w;
    p.h[4] = (__bf16)b.x; p.h[5] = (__bf16)b.y; p.h[6] = (__bf16)b.z; p.h[7] = (__bf16)b.w;
    *reinterpret_cast<uint4*>(xb + i) = p.u;
}

// ---------------------------------------------------------------------------
// Epilogue: D tile (o-major rows in VGPRs, m per lane) + bias -> out[m, o].
// Per lane: 8 contiguous floats along o -> two non-temporal float4 stores
// (output is write-once; keep it out of the L2 that holds W/x).
// ---------------------------------------------------------------------------
__device__ __forceinline__ void store_tile(
    v8f acc, int o_start, int m,
    const float* __restrict__ bias, float* __restrict__ out)
{
    const v4f b0 = *reinterpret_cast<const v4f*>(bias + o_start);
    const v4f b1 = *reinterpret_cast<const v4f*>(bias + o_start + 4);
    v4f r0, r1;
    r0.x = acc[0] + b0.x; r0.y = acc[1] + b0.y; r0.z = acc[2] + b0.z; r0.w = acc[3] + b0.w;
    r1.x = acc[4] + b1.x; r1.y = acc[5] + b1.y; r1.z = acc[6] + b1.z; r1.w = acc[7] + b1.w;
    float* op = out + (size_t)m * OUT_F + o_start;
    __builtin_nontemporal_store(r0, reinterpret_cast<v4f*>(op));
    __builtin_nontemporal_store(r1, reinterpret_cast<v4f*>(op + 4));
}

// ---------------------------------------------------------------------------
// Kernel 3: SWMMAC main loop.
//   wave tile:  64(o) x 32(m)  -> 4 A frags (reused 2x) x 2 B frags (reused 4x)
//   block tile: 128(o) x 128(m), 8 waves as 2(o) x 4(m)
// ---------------------------------------------------------------------------
__global__ __launch_bounds__(256) void k_spgemm(
    const __bf16* __restrict__ wv, const uint8_t* __restrict__ widx,
    const __bf16* __restrict__ xb, const float* __restrict__ bias,
    float* __restrict__ out)
{
    const int lane = threadIdx.x & 31;
    const int wave = threadIdx.x >> 5;
    const int hi   = lane >> 4;          // lane half selects K-chunk
    const int lo   = lane & 15;          // A row / B column within tile

    const int o_base = blockIdx.x * 128 + (wave & 1) * 64;
    const int m_base = blockIdx.y * 128 + (wave >> 1) * 32;

    // A stored 16x32 (bf16): lanes 0-15 hold stored K' {0..7,16..23},
    // lanes 16-31 hold {8..15,24..31}  -> fold hi*8 into the base pointer.
    // Index dword: lanes 0-15 cover original cols 0..31, lanes 16-31 cols 32..63.
    const __bf16*  ap[4];
    const uint8_t* ip[4];
#pragma unroll
    for (int j = 0; j < 4; ++j) {
        const int row = o_base + j * 16 + lo;
        ap[j] = wv   + (size_t)row * KSTOR + hi * 8;
        ip[j] = widx + (size_t)row * KIDXB + hi * 4;
    }
    // B 64x16: V0-7 lanes0-15 K0-15 / lanes16-31 K16-31; V8-15 K32-47 / K48-63.
    const __bf16* bp[2];
#pragma unroll
    for (int j = 0; j < 2; ++j) {
        const int m = m_base + j * 16 + lo;
        bp[j] = xb + (size_t)m * IN_F + hi * 16;
    }

    v8f acc[4][2];
#pragma unroll
    for (int j = 0; j < 4; ++j)
#pragma unroll
        for (int i = 0; i < 2; ++i) acc[j][i] = (v8f){};

    for (int k0 = 0; k0 < IN_F; k0 += 64) {
        const int kk = k0 >> 1;          // stored-K offset for this tile

        union { uint4 q[2]; v16bf v; } A[4];
        union { uint4 q[4]; v32bf v; } B[2];
        int idx[4];

#pragma unroll
        for (int j = 0; j < 4; ++j) {
            A[j].q[0] = *reinterpret_cast<const uint4*>(ap[j] + kk);
            A[j].q[1] = *reinterpret_cast<const uint4*>(ap[j] + kk + 16);
            idx[j]    = *reinterpret_cast<const int*>(ip[j] + (k0 >> 3));
        }
#pragma unroll
        for (int j = 0; j < 2; ++j) {
            B[j].q[0] = *reinterpret_cast<const uint4*>(bp[j] + k0);
            B[j].q[1] = *reinterpret_cast<const uint4*>(bp[j] + k0 + 8);
            B[j].q[2] = *reinterpret_cast<const uint4*>(bp[j] + k0 + 32);
            B[j].q[3] = *reinterpret_cast<const uint4*>(bp[j] + k0 + 40);
        }

        // Prefetch next K-tile streams into GL2 (global_prefetch_b8).
        __builtin_prefetch(bp[0] + k0 + 64, 0, 1);
        __builtin_prefetch(bp[1] + k0 + 64, 0, 1);
        __builtin_prefetch(ap[0] + kk + 32, 0, 1);
        __builtin_prefetch(ap[2] + kk + 32, 0, 1);

        // D = A x B + C  (2:4 sparse A, index in SRC2): 8 swmmac per K-step.
#pragma unroll
        for (int j = 0; j < 4; ++j) {
            acc[j][0] = __builtin_amdgcn_swmmac_f32_16x16x64_bf16(
                            false, A[j].v, false, B[0].v, acc[j][0], idx[j], false, false);
            acc[j][1] = __builtin_amdgcn_swmmac_f32_16x16x64_bf16(
                            false, A[j].v, false, B[1].v, acc[j][1], idx[j], false, false);
        }
    }

#pragma unroll
    for (int j = 0; j < 4; ++j)
#pragma unroll
        for (int i = 0; i < 2; ++i)
            store_tile(acc[j][i], o_base + j * 16 + hi * 8,
                       m_base + i * 16 + lo, bias, out);
}

// ---------------------------------------------------------------------------
extern "C" void kernel_launch(void* const* d_in, const int* in_sizes, int n_in,
                              void* d_out, int out_size, void* d_ws, size_t ws_size,
                              hipStream_t stream)
{
    const float*   x    = (const float*)  d_in[0];   // [4096, 4096]
    const float*   w    = (const float*)  d_in[1];   // [16384, 4096]
    const uint8_t* mask = (const uint8_t*)d_in[2];   // [16384, 4096] bool
    const float*   bias = (const float*)  d_in[3];   // [16384]
    float*         out  = (float*)        d_out;     // [4096, 16384]

    // Workspace layout: packed W values (64 MB) | indices (8 MB) | x bf16 (32 MB)
    uint8_t* ws   = (uint8_t*)d_ws;
    __bf16*  wv   = (__bf16*)(ws);
    uint8_t* widx = ws + (size_t)OUT_F * KSTOR * 2;
    __bf16*  xb   = (__bf16*)(ws + (size_t)OUT_F * KSTOR * 2 + (size_t)OUT_F * KIDXB);

    k_compress_w<<<(OUT_F * KIDXB) / 256, 256, 0, stream>>>(w, mask, wv, widx);

    const long nx = (long)MROWS * IN_F;
    k_cvt_x<<<(int)(nx / 8 / 256), 256, 0, stream>>>(x, xb, nx);

    dim3 grid(OUT_F / 128, MROWS / 128);
    k_spgemm<<<grid, dim3(256), 0, stream>>>(wv, widx, xb, bias, out);
}